// static_heto_graph_45732811768428
// MI455X (gfx1250) — compile-verified
//
#include <hip/hip_runtime.h>
#include <math.h>

typedef float v2f __attribute__((ext_vector_type(2)));
typedef float v8f __attribute__((ext_vector_type(8)));

#define H_DIM 128
#define N_DOC 16384
#define N_WORD_NODES 100000
#define N_TOPIC_NODES 1600
#define E_WD 1000000
#define E_TD 200000
#define NB 32
#define DOCS_PER_GRAPH 512

// ---------------- workspace layout (floats) ----------------
#define OFF_OUTDEG_WD 0
#define OFF_INDEG_WD  (OFF_OUTDEG_WD + N_WORD_NODES)
#define OFF_OUTDEG_TD (OFF_INDEG_WD + N_DOC)
#define OFF_INDEG_TD  (OFF_OUTDEG_TD + N_TOPIC_NODES)
#define OFF_AGG_WD    (OFF_INDEG_TD + N_DOC)
#define OFF_AGG_TD    (OFF_AGG_WD + (size_t)N_DOC * H_DIM)
#define OFF_POOL      (OFF_AGG_TD + (size_t)N_DOC * H_DIM)
#define WS_FLOATS     (OFF_POOL + (size_t)NB * H_DIM)

// ---------------- kernel 0: zero workspace ----------------
__global__ void k_zero(float* __restrict__ ws, size_t n) {
    size_t i = (size_t)blockIdx.x * blockDim.x + threadIdx.x;
    size_t stride = (size_t)gridDim.x * blockDim.x;
    for (; i < n; i += stride) ws[i] = 0.0f;
}

// ---------------- kernel 1: degree histograms -------------
__global__ void k_degrees(const int* __restrict__ src, const int* __restrict__ dst,
                          float* __restrict__ outdeg, float* __restrict__ indeg, int n) {
    int i = blockIdx.x * blockDim.x + threadIdx.x;
    if (i < n) {
        atomicAdd(&outdeg[src[i]], 1.0f);
        atomicAdd(&indeg[dst[i]], 1.0f);
    }
}

// ---------------- kernel 2: edge scatter ------------------
// One wave32 per edge; each lane moves float4 (lane*4 .. lane*4+3) of the
// 128-wide feature row: one GLOBAL_LOAD_B128 + 4 GLOBAL_ATOMIC_ADD_F32.
__global__ __launch_bounds__(256) void k_scatter(
    const int* __restrict__ src, const int* __restrict__ dst,
    const int* __restrict__ node_ids, const float* __restrict__ embeds,
    const float* __restrict__ outdeg, float* __restrict__ agg, int n_edges) {
    int e = blockIdx.x * 8 + (threadIdx.x >> 5);
    int lane = threadIdx.x & 31;
    if (e >= n_edges) return;
    int s = src[e];
    int d = dst[e];
    float dg = outdeg[s];
    float scale = rsqrtf(dg < 1.0f ? 1.0f : dg);
    const float4* row = (const float4*)(embeds + (size_t)node_ids[s] * H_DIM);
    float4 v = row[lane];
    float* out = agg + (size_t)d * H_DIM + lane * 4;
    atomicAdd(out + 0, v.x * scale);
    atomicAdd(out + 1, v.y * scale);
    atomicAdd(out + 2, v.z * scale);
    atomicAdd(out + 3, v.w * scale);
}

// ---------------- kernel 3: WMMA GEMM + scale + pool ------
// One wave per 16-doc tile: out[16,128] = agg[16,128] @ W[128,128], then
// rows scaled by rsqrt(indeg), + bias, summed over rows into per-graph pool.
// A-frag (16x4 f32): lane L holds m=L&15, VGPRs = K 2*(L>>4), 2*(L>>4)+1.
// B-frag (4x16 f32): lane L holds n=L&15, VGPRs = K 2*(L>>4), 2*(L>>4)+1.
// C/D 16x16 f32: VGPR r, lane L -> row r+8*(L>=16), col L&15.
__device__ __forceinline__ void gemm_rel_pool(
    const float* __restrict__ A, const float* __restrict__ W,
    const float* __restrict__ bias, const float* __restrict__ indeg,
    int doc0, int lan, int hi, float* pooled) {
    v8f acc[8];
#pragma unroll
    for (int nt = 0; nt < 8; ++nt) acc[nt] = (v8f){0, 0, 0, 0, 0, 0, 0, 0};
    const float* Arow = A + (size_t)(doc0 + lan) * H_DIM;
    for (int k = 0; k < H_DIM; k += 4) {
        v2f a;
        a.x = Arow[k + 2 * hi];
        a.y = Arow[k + 2 * hi + 1];
#pragma unroll
        for (int nt = 0; nt < 8; ++nt) {
            int n = nt * 16 + lan;
            v2f b;
            b.x = W[(k + 2 * hi) * H_DIM + n];
            b.y = W[(k + 2 * hi + 1) * H_DIM + n];
            acc[nt] = __builtin_amdgcn_wmma_f32_16x16x4_f32(
                false, a, false, b, (short)0, acc[nt], false, false);
        }
    }
    float s[8];
#pragma unroll
    for (int r = 0; r < 8; ++r) {
        float dg = indeg[doc0 + r + 8 * hi];
        s[r] = rsqrtf(dg < 1.0f ? 1.0f : dg);
    }
#pragma unroll
    for (int nt = 0; nt < 8; ++nt) {
        int n = nt * 16 + lan;
        float t = 8.0f * bias[n];  // 8 rows per lane-half each get +bias[n]
#pragma unroll
        for (int r = 0; r < 8; ++r) t += acc[nt][r] * s[r];
        pooled[nt] += t;
    }
}

__global__ __launch_bounds__(256) void k_gemm_pool(
    const float* __restrict__ agg_wd, const float* __restrict__ agg_td,
    const float* __restrict__ indeg_wd, const float* __restrict__ indeg_td,
    const float* __restrict__ W_wd, const float* __restrict__ b_wd,
    const float* __restrict__ W_td, const float* __restrict__ b_td,
    float* __restrict__ pool) {
    int tile = blockIdx.x * 8 + (threadIdx.x >> 5);  // 1024 tiles of 16 docs
    int lane = threadIdx.x & 31;
    int hi = lane >> 4;
    int lan = lane & 15;
    int doc0 = tile * 16;

    float pooled[8];
#pragma unroll
    for (int nt = 0; nt < 8; ++nt) pooled[nt] = 0.0f;

    gemm_rel_pool(agg_wd, W_wd, b_wd, indeg_wd, doc0, lan, hi, pooled);
    gemm_rel_pool(agg_td, W_td, b_td, indeg_td, doc0, lan, hi, pooled);

    int g = doc0 / DOCS_PER_GRAPH;  // each 16-doc tile lies in one graph
#pragma unroll
    for (int nt = 0; nt < 8; ++nt)
        atomicAdd(&pool[g * H_DIM + nt * 16 + lan], pooled[nt]);
}

// ---------------- kernel 4: head (logits, sigmoid, BCE) ---
__global__ __launch_bounds__(256) void k_head(
    const float* __restrict__ pool, const float* __restrict__ W_out,
    const float* __restrict__ b_out, const float* __restrict__ y,
    float* __restrict__ out) {
    __shared__ float lds_loss[NB];
    int wave = threadIdx.x >> 5;
    int lane = threadIdx.x & 31;
    for (int g = wave; g < NB; g += 8) {
        float4 p = ((const float4*)(pool + g * H_DIM))[lane];
        float4 w = ((const float4*)W_out)[lane];
        float part = p.x * w.x + p.y * w.y + p.z * w.z + p.w * w.w;
        for (int off = 16; off > 0; off >>= 1)
            part += __shfl_down(part, off, 32);
        if (lane == 0) {
            float x = part * (1.0f / (float)DOCS_PER_GRAPH) + b_out[0];
            out[1 + g] = 1.0f / (1.0f + expf(-x));
            lds_loss[g] = fmaxf(x, 0.0f) - x * y[g] + log1pf(expf(-fabsf(x)));
        }
    }
    __syncthreads();
    if (threadIdx.x == 0) {
        float sum = 0.0f;
        for (int g = 0; g < NB; ++g) sum += lds_loss[g];
        out[0] = sum * (1.0f / (float)NB);
    }
}

// ---------------- launch ----------------------------------
extern "C" void kernel_launch(void* const* d_in, const int* in_sizes, int n_in,
                              void* d_out, int out_size, void* d_ws, size_t ws_size,
                              hipStream_t stream) {
    const int*   word_ids     = (const int*)d_in[0];
    const int*   topic_ids    = (const int*)d_in[1];
    const int*   wd_src       = (const int*)d_in[2];
    const int*   wd_dst       = (const int*)d_in[3];
    const int*   td_src       = (const int*)d_in[4];
    const int*   td_dst       = (const int*)d_in[5];
    const float* y_data       = (const float*)d_in[6];
    const float* word_embeds  = (const float*)d_in[7];
    const float* topic_embeds = (const float*)d_in[8];
    const float* W_wd         = (const float*)d_in[9];
    const float* b_wd         = (const float*)d_in[10];
    const float* W_td         = (const float*)d_in[11];
    const float* b_td         = (const float*)d_in[12];
    const float* W_out        = (const float*)d_in[13];
    const float* b_out        = (const float*)d_in[14];
    float* out = (float*)d_out;

    float* ws        = (float*)d_ws;
    float* outdeg_wd = ws + OFF_OUTDEG_WD;
    float* indeg_wd  = ws + OFF_INDEG_WD;
    float* outdeg_td = ws + OFF_OUTDEG_TD;
    float* indeg_td  = ws + OFF_INDEG_TD;
    float* agg_wd    = ws + OFF_AGG_WD;
    float* agg_td    = ws + OFF_AGG_TD;
    float* pool      = ws + OFF_POOL;

    k_zero<<<2048, 256, 0, stream>>>(ws, WS_FLOATS);

    k_degrees<<<(E_WD + 255) / 256, 256, 0, stream>>>(wd_src, wd_dst, outdeg_wd, indeg_wd, E_WD);
    k_degrees<<<(E_TD + 255) / 256, 256, 0, stream>>>(td_src, td_dst, outdeg_td, indeg_td, E_TD);

    k_scatter<<<(E_WD + 7) / 8, 256, 0, stream>>>(wd_src, wd_dst, word_ids, word_embeds,
                                                  outdeg_wd, agg_wd, E_WD);
    k_scatter<<<(E_TD + 7) / 8, 256, 0, stream>>>(td_src, td_dst, topic_ids, topic_embeds,
                                                  outdeg_td, agg_td, E_TD);

    k_gemm_pool<<<(N_DOC / 16) / 8, 256, 0, stream>>>(agg_wd, agg_td, indeg_wd, indeg_td,
                                                      W_wd, b_wd, W_td, b_td, pool);

    k_head<<<1, 256, 0, stream>>>(pool, W_out, b_out, y_data, out);
}